// MPNN_9388798509092
// MI455X (gfx1250) — compile-verified
//
#include <hip/hip_runtime.h>
#include <hip/hip_bf16.h>

// MPNN: B=4096 graphs, N=64 nodes, D=128, L=3 layers, fp32 throughout.
// One workgroup (256 threads = 8 wave32) per graph; all intermediates in LDS;
// both GEMMs per layer run on v_wmma_f32_16x16x4_f32.

typedef __attribute__((ext_vector_type(2))) float v2f;
typedef __attribute__((ext_vector_type(8))) float v8f;

#define NB   4096
#define NN   64
#define ND   128

#define HS 132   // h row stride (dwords): %64==4 -> conflict-free row-strided reads
#define ZS 144   // z row stride: 2*144%64==32 -> disjoint bank sets for half-waves
#define WS 132   // W row stride
#define AS 68    // adj row stride

__global__ __launch_bounds__(256)
void mpnn_wmma_kernel(const int* __restrict__ fps,
                      const float* __restrict__ adj,
                      const float* __restrict__ emb,
                      const float* __restrict__ Wg,
                      const float* __restrict__ bg,
                      float* __restrict__ out)
{
    __shared__ __align__(16) float h_s[NN * HS];       // 33.0 KB
    __shared__ __align__(16) float z_s[NN * ZS];       // 36.0 KB
    __shared__ __align__(16) float w_s[ND * WS];       // 66.0 KB
    __shared__ __align__(16) float a_s[NN * AS];       // 17.0 KB
    __shared__ float b_s[ND];
    __shared__ int   id_s[NN];

    const int g    = blockIdx.x;
    const int tid  = threadIdx.x;
    const int lane = tid & 31;
    const int wave = tid >> 5;                 // 0..7

    const int m0   = (wave >> 1) * 16;         // output row-tile base: 0,16,32,48
    const int n0   = (wave & 1) * 64;          // output column-half base: 0 or 64
    const int lc   = lane & 15;                // lane's column within a 16-wide tile
    const int koff = (lane >> 4) * 2;          // K offset for A/B frags (0 or 2)
    const int rofs = (lane >> 4) * 8;          // C/D row offset for regs (0 or 8)

    if (tid < NN) id_s[tid] = fps[g * NN + tid];

    // Stage adjacency block (64x64) into padded LDS (float4 granularity).
    {
        const float4* ag = (const float4*)(adj + (size_t)g * NN * NN);
        for (int i = tid; i < NN * (NN / 4); i += 256) {
            int row = i >> 4, c4 = i & 15;
            *(float4*)&a_s[row * AS + c4 * 4] = ag[row * 16 + c4];
        }
    }
    __syncthreads();

    // Gather node embeddings: h = emb[fps] (row per node, 128 floats).
    for (int i = tid; i < NN * (ND / 4); i += 256) {
        int row = i >> 5, c4 = i & 31;
        const float4* er = (const float4*)(emb + (size_t)id_s[row] * ND);
        *(float4*)&h_s[row * HS + c4 * 4] = er[c4];
    }

    for (int l = 0; l < 3; ++l) {
        __syncthreads();   // h complete; previous layer's LDS reads finished

        // Stage W[l] (128x128) and b[l] into LDS.
        const float4* wg4 = (const float4*)(Wg + (size_t)l * ND * ND);
        for (int i = tid; i < ND * (ND / 4); i += 256) {
            int row = i >> 5, c4 = i & 31;
            *(float4*)&w_s[row * WS + c4 * 4] = wg4[row * 32 + c4];
        }
        if (tid < ND) b_s[tid] = bg[l * ND + tid];
        __syncthreads();

        // ---- GEMM 1: z = relu(h @ W^T + b) ----
        // A = h[16x128 strip], B[kd][col] = W[n0+col][k+kd]  (W^T access)
        v8f acc[4];
        {
            v8f zero = {};
            #pragma unroll
            for (int t = 0; t < 4; ++t) acc[t] = zero;
        }
        const int arow = m0 + lc;
        for (int k = 0; k < ND; k += 4) {
            v2f a = *(const v2f*)&h_s[arow * HS + k + koff];
            #pragma unroll
            for (int t = 0; t < 4; ++t) {
                v2f b = *(const v2f*)&w_s[(n0 + t * 16 + lc) * WS + k + koff];
                acc[t] = __builtin_amdgcn_wmma_f32_16x16x4_f32(
                    false, a, false, b, (short)0, acc[t], false, false);
            }
        }
        // bias + relu, store to z
        #pragma unroll
        for (int t = 0; t < 4; ++t) {
            int col = n0 + t * 16 + lc;
            float bv = b_s[col];
            #pragma unroll
            for (int r = 0; r < 8; ++r) {
                float v = acc[t][r] + bv;
                z_s[(m0 + rofs + r) * ZS + col] = v > 0.f ? v : 0.f;
            }
        }
        __syncthreads();   // z complete across all waves

        // ---- GEMM 2: h = z + adj @ z ----
        // A = adj[16x64 strip], B[kd][col] = z[k+kd][n0+col]
        {
            v8f zero = {};
            #pragma unroll
            for (int t = 0; t < 4; ++t) acc[t] = zero;
        }
        for (int k = 0; k < NN; k += 4) {
            v2f a = *(const v2f*)&a_s[arow * AS + k + koff];
            int r0 = (k + koff) * ZS;
            int r1 = r0 + ZS;
            #pragma unroll
            for (int t = 0; t < 4; ++t) {
                int col = n0 + t * 16 + lc;
                v2f b;
                b.x = z_s[r0 + col];
                b.y = z_s[r1 + col];
                acc[t] = __builtin_amdgcn_wmma_f32_16x16x4_f32(
                    false, a, false, b, (short)0, acc[t], false, false);
            }
        }
        // h = z + (adj @ z)
        #pragma unroll
        for (int t = 0; t < 4; ++t) {
            int col = n0 + t * 16 + lc;
            #pragma unroll
            for (int r = 0; r < 8; ++r) {
                int row = m0 + rofs + r;
                h_s[row * HS + col] = z_s[row * ZS + col] + acc[t][r];
            }
        }
    }

    __syncthreads();
    // Sum-pool over nodes: out[g][d] = sum_n h[n][d]
    if (tid < ND) {
        float s = 0.f;
        #pragma unroll 4
        for (int n = 0; n < NN; ++n) s += h_s[n * HS + tid];
        out[(size_t)g * ND + tid] = s;
    }
}

extern "C" void kernel_launch(void* const* d_in, const int* in_sizes, int n_in,
                              void* d_out, int out_size, void* d_ws, size_t ws_size,
                              hipStream_t stream) {
    const int*   fps = (const int*)d_in[0];
    const float* adj = (const float*)d_in[1];
    const float* emb = (const float*)d_in[2];
    const float* W   = (const float*)d_in[3];
    const float* b   = (const float*)d_in[4];
    float* out = (float*)d_out;

    mpnn_wmma_kernel<<<NB, 256, 0, stream>>>(fps, adj, emb, W, b, out);
}